// TemporalPointNet_33234456936971
// MI455X (gfx1250) — compile-verified
//
#include <hip/hip_runtime.h>
#include <hip/hip_bf16.h>

typedef __attribute__((ext_vector_type(2))) float f32x2;
typedef __attribute__((ext_vector_type(8))) float f32x8;

// ---------------------------------------------------------------------------
// Farthest point sampling: one block per batch, whole cloud cached in LDS.
// Emits incoming `far` (starting at 0), then updates dist with that point and
// takes argmax (lowest index on ties) — matches the jax scan exactly.
// ---------------------------------------------------------------------------
__global__ void fps_kernel(const float* __restrict__ pts, int N, int stride,
                           int npoint, int* __restrict__ out) {
  const int b = blockIdx.x;
  const int tid = threadIdx.x;
  const int bd = blockDim.x;
  extern __shared__ float sm[];
  float* px = sm;
  float* py = px + N;
  float* pz = py + N;
  float* dist = pz + N;
  float* rv = dist + N;
  int* ri = (int*)(rv + bd);
  __shared__ int far_s;
  const float* base = pts + (size_t)b * N * stride;
  for (int n = tid; n < N; n += bd) {
    px[n] = base[(size_t)n * stride + 0];
    py[n] = base[(size_t)n * stride + 1];
    pz[n] = base[(size_t)n * stride + 2];
    dist[n] = 1e10f;
  }
  if (tid == 0) far_s = 0;
  __syncthreads();
  for (int it = 0; it < npoint; ++it) {
    const int far = far_s;
    if (tid == 0) out[b * npoint + it] = far;
    const float cx = px[far], cy = py[far], cz = pz[far];
    float best = -1.0f;
    int bi = 0x7fffffff;
    for (int n = tid; n < N; n += bd) {
      const float dx = px[n] - cx, dy = py[n] - cy, dz = pz[n] - cz;
      const float d = dx * dx + dy * dy + dz * dz;
      const float dd = fminf(dist[n], d);
      dist[n] = dd;
      if (dd > best) { best = dd; bi = n; }
    }
    rv[tid] = best; ri[tid] = bi;
    __syncthreads();
    for (int s = bd >> 1; s > 0; s >>= 1) {
      if (tid < s) {
        const float ov = rv[tid + s]; const int oi = ri[tid + s];
        if (ov > rv[tid] || (ov == rv[tid] && oi < ri[tid])) { rv[tid] = ov; ri[tid] = oi; }
      }
      __syncthreads();
    }
    if (tid == 0) far_s = ri[0];
    __syncthreads();
  }
}

// gather xyz coordinates of sampled centers: out[(b*S+s)*3+c]
__global__ void gather_xyz_kernel(const float* __restrict__ pts, int N, int stride,
                                  const int* __restrict__ fidx, int S,
                                  float* __restrict__ out, int total) {
  const int t = blockIdx.x * blockDim.x + threadIdx.x;
  if (t >= total) return;
  const int c = t % 3;
  const int bs = t / 3;
  const int b = bs / S;
  const int n = fidx[bs];
  out[t] = pts[((size_t)b * N + n) * stride + c];
}

// Ball query: first `nsample` indices (ascending) with d^2 <= r^2, padded with
// the first hit. One thread per center.
__global__ void query_ball_kernel(const float* __restrict__ pts, int N, int stride,
                                  const float* __restrict__ new_xyz, float r2,
                                  int nsample, int* __restrict__ gidx, int S, int total) {
  const int g = blockIdx.x * blockDim.x + threadIdx.x;
  if (g >= total) return;
  const int b = g / S;
  const float cx = new_xyz[(size_t)g * 3 + 0];
  const float cy = new_xyz[(size_t)g * 3 + 1];
  const float cz = new_xyz[(size_t)g * 3 + 2];
  const float* base = pts + (size_t)b * N * stride;
  int* out = gidx + (size_t)g * nsample;
  int cnt = 0;
  int first = 0;
  for (int n = 0; n < N && cnt < nsample; ++n) {
    const float dx = base[(size_t)n * stride + 0] - cx;
    const float dy = base[(size_t)n * stride + 1] - cy;
    const float dz = base[(size_t)n * stride + 2] - cz;
    if (dx * dx + dy * dy + dz * dz <= r2) {
      if (cnt == 0) first = n;
      out[cnt++] = n;
    }
  }
  for (; cnt < nsample; ++cnt) out[cnt] = first;
}

// Build grouped input rows with padded row stride CP (pad columns zeroed):
// X[t][0..2] = grouped_xyz - center, X[t][3..3+F) = feat, X[t][3+F..CP) = 0.
__global__ void build_points_kernel(const float* __restrict__ pts, int N, int pstride,
                                    const float* __restrict__ feat, int fstride, int F,
                                    const float* __restrict__ new_xyz,
                                    const int* __restrict__ gidx,
                                    int S, int nsample, int CP,
                                    float* __restrict__ X, int total) {
  const int t = blockIdx.x * blockDim.x + threadIdx.x;
  if (t >= total) return;
  const int bs = t / nsample;
  const int b = bs / S;
  const int n = gidx[t];
  float* xr = X + (size_t)t * CP;
  const float* p = pts + ((size_t)b * N + n) * pstride;
  const float* cc = new_xyz + (size_t)bs * 3;
  xr[0] = p[0] - cc[0];
  xr[1] = p[1] - cc[1];
  xr[2] = p[2] - cc[2];
  const float* f = feat + ((size_t)b * N + n) * fstride;
  for (int c = 0; c < F; ++c) xr[3 + c] = f[c];
  for (int c = 3 + F; c < CP; ++c) xr[c] = 0.0f;
}

// group_all concat (no centering), padded stride CP:
// X[m][c] = c<3 ? xyz : (c<3+F ? feat : 0)
__global__ void concat_all_kernel(const float* __restrict__ xyz,
                                  const float* __restrict__ feat, int F, int CP,
                                  float* __restrict__ X, int total) {
  const int t = blockIdx.x * blockDim.x + threadIdx.x;
  if (t >= total) return;
  const int m = t / CP;
  const int c = t % CP;
  float v = 0.0f;
  if (c < 3) v = xyz[(size_t)m * 3 + c];
  else if (c < 3 + F) v = feat[(size_t)m * F + (c - 3)];
  X[t] = v;
}

// Zero-pad weight rows: Wp[n*KPAD+k] = (k<K) ? W[n*K+k] : 0
__global__ void pack_weights_kernel(const float* __restrict__ W, int K, int KPAD,
                                    float* __restrict__ Wp, int total) {
  const int t = blockIdx.x * blockDim.x + threadIdx.x;
  if (t >= total) return;
  const int n = t / KPAD;
  const int k = t % KPAD;
  Wp[t] = (k < K) ? W[(size_t)n * K + k] : 0.0f;
}

// ---------------------------------------------------------------------------
// fp32 WMMA GEMM: Y[M,N] = X[M,KPAD] * W[N,KPAD]^T with V_WMMA_F32_16X16X4_F32.
// KPAD is a compile-time multiple of 4 -> fully unrolled, unpredicated b64
// loads. Each wave computes a 16x32 output (two accumulators sharing the A
// fragment); 4 waves (128 threads) per block cover 64 rows.
// A frag (16x4 f32): lane<16 -> M=lane, {K0,K1}; lane>=16 -> {K2,K3}.
// B frag (4x16 f32): col = lane&15, rows kc*4 + 2*(lane>>4) + {0,1}.
// C/D: vgpr v -> row m0 + v + 8*(lane>>4), col n0 + (lane&15).
// ---------------------------------------------------------------------------
template <int KPAD>
__global__ __launch_bounds__(128) void gemm_wmma_f32(const float* __restrict__ X,
                                                     const float* __restrict__ W,
                                                     float* __restrict__ Y, int N) {
  const int lane = threadIdx.x & 31;
  const int wave = threadIdx.x >> 5;
  const int m0 = (blockIdx.x * 4 + wave) * 16;
  const int n0 = blockIdx.y * 32;
  const int half = lane >> 4;
  const int l = lane & 15;
  const f32x2* xrow = (const f32x2*)(X + (size_t)(m0 + l) * KPAD) + half;
  const f32x2* wrow0 = (const f32x2*)(W + (size_t)(n0 + l) * KPAD) + half;
  const f32x2* wrow1 = (const f32x2*)(W + (size_t)(n0 + 16 + l) * KPAD) + half;
  f32x8 acc0 = {0.f, 0.f, 0.f, 0.f, 0.f, 0.f, 0.f, 0.f};
  f32x8 acc1 = {0.f, 0.f, 0.f, 0.f, 0.f, 0.f, 0.f, 0.f};
#pragma unroll
  for (int kc = 0; kc < KPAD / 4; ++kc) {
    const f32x2 a = xrow[kc * 2];
    const f32x2 b0 = wrow0[kc * 2];
    const f32x2 b1 = wrow1[kc * 2];
    acc0 = __builtin_amdgcn_wmma_f32_16x16x4_f32(false, a, false, b0,
                                                 (short)0, acc0, false, false);
    acc1 = __builtin_amdgcn_wmma_f32_16x16x4_f32(false, a, false, b1,
                                                 (short)0, acc1, false, false);
  }
  float* yb0 = Y + (size_t)(m0 + half * 8) * N + n0 + l;
  float* yb1 = yb0 + 16;
#pragma unroll
  for (int v = 0; v < 8; ++v) {
    yb0[(size_t)v * N] = acc0[v];
    yb1[(size_t)v * N] = acc1[v];
  }
}

static void launch_gemm(const float* X, const float* W, float* Y,
                        int M, int N, int KPAD, hipStream_t s) {
  dim3 grid(M / 64, N / 32);
  switch (KPAD) {
    case 4:   gemm_wmma_f32<4><<<grid, 128, 0, s>>>(X, W, Y, N); break;
    case 64:  gemm_wmma_f32<64><<<grid, 128, 0, s>>>(X, W, Y, N); break;
    case 128: gemm_wmma_f32<128><<<grid, 128, 0, s>>>(X, W, Y, N); break;
    case 132: gemm_wmma_f32<132><<<grid, 128, 0, s>>>(X, W, Y, N); break;
    case 256: gemm_wmma_f32<256><<<grid, 128, 0, s>>>(X, W, Y, N); break;
    case 260: gemm_wmma_f32<260><<<grid, 128, 0, s>>>(X, W, Y, N); break;
    case 512: gemm_wmma_f32<512><<<grid, 128, 0, s>>>(X, W, Y, N); break;
    default: break;
  }
}

// Per-channel mean/var over all M rows -> scale = g*rsqrt(var+eps),
// shift = b - mean*scale. One block per channel.
__global__ void bn_stats_kernel(const float* __restrict__ Y, int M, int O,
                                const float* __restrict__ g, const float* __restrict__ b,
                                float* __restrict__ scale, float* __restrict__ shift) {
  const int o = blockIdx.x;
  const int tid = threadIdx.x;
  double s = 0.0, sq = 0.0;
  for (int m = tid; m < M; m += blockDim.x) {
    const float v = Y[(size_t)m * O + o];
    s += (double)v;
    sq += (double)v * (double)v;
  }
  __shared__ double sv[256];
  __shared__ double sqv[256];
  sv[tid] = s; sqv[tid] = sq;
  __syncthreads();
  for (int r = 128; r > 0; r >>= 1) {
    if (tid < r) { sv[tid] += sv[tid + r]; sqv[tid] += sqv[tid + r]; }
    __syncthreads();
  }
  if (tid == 0) {
    const double mean = sv[0] / (double)M;
    double var = sqv[0] / (double)M - mean * mean;
    if (var < 0.0) var = 0.0;
    const float sc = g[o] * rsqrtf((float)var + 1e-5f);
    scale[o] = sc;
    shift[o] = b[o] - (float)mean * sc;
  }
}

__global__ void bn_relu_kernel(float* __restrict__ Y, const float* __restrict__ scale,
                               const float* __restrict__ shift, int O, int total) {
  const int t = blockIdx.x * blockDim.x + threadIdx.x;
  if (t >= total) return;
  const int o = t % O;
  Y[t] = fmaxf(Y[t] * scale[o] + shift[o], 0.0f);
}

// max over the sample axis: out[ms][o] = max_k Y[(ms*Ks+k)*O + o]
__global__ void maxpool_kernel(const float* __restrict__ Y, int Ks, int O,
                               float* __restrict__ out, int total) {
  const int t = blockIdx.x * blockDim.x + threadIdx.x;
  if (t >= total) return;
  const int ms = t / O;
  const int o = t % O;
  float m = -1e30f;
  for (int k = 0; k < Ks; ++k) m = fmaxf(m, Y[((size_t)ms * Ks + k) * O + o]);
  out[t] = m;
}

// ---------------------------------------------------------------------------
extern "C" void kernel_launch(void* const* d_in, const int* in_sizes, int n_in,
                              void* d_out, int out_size, void* d_ws, size_t ws_size,
                              hipStream_t stream) {
  (void)in_sizes; (void)n_in; (void)out_size; (void)ws_size;
  const float* xyz_time = (const float*)d_in[0];  // (2,4,4096,4) -> (8,4096,4)
  auto Wgt = [&](int s, int j) { return (const float*)d_in[1 + (s - 1) * 9 + j * 3 + 0]; };
  auto Gam = [&](int s, int j) { return (const float*)d_in[1 + (s - 1) * 9 + j * 3 + 1]; };
  auto Bet = [&](int s, int j) { return (const float*)d_in[1 + (s - 1) * 9 + j * 3 + 2]; };

  const int B = 8;          // B*T
  const int N1 = 4096, S1 = 512, NS1 = 32;
  const int S2 = 128, NS2 = 64;
  const int M1 = B * S1 * NS1;   // 131072
  const int M2 = B * S2 * NS2;   // 65536
  const int M3 = B * S2;         // 1024

  char* ws = (char*)d_ws;
  size_t off = 0;
  auto take = [&](size_t bytes) -> void* {
    void* p = ws + off;
    off += (bytes + 255) & ~(size_t)255;
    return p;
  };
  float* bufA  = (float*)take((size_t)M1 * 128 * 4);   // 64 MiB ping
  float* bufB  = (float*)take((size_t)M1 * 128 * 4);   // 64 MiB pong
  float* X1    = (float*)take((size_t)M1 * 4 * 4);
  int*   fidx1 = (int*)  take((size_t)B * S1 * 4);
  float* nxyz1 = (float*)take((size_t)B * S1 * 3 * 4);
  int*   gidx1 = (int*)  take((size_t)B * S1 * NS1 * 4);
  float* feat1 = (float*)take((size_t)B * S1 * 128 * 4);
  int*   fidx2 = (int*)  take((size_t)B * S2 * 4);
  float* nxyz2 = (float*)take((size_t)B * S2 * 3 * 4);
  int*   gidx2 = (int*)  take((size_t)B * S2 * NS2 * 4);
  float* feat2 = (float*)take((size_t)M3 * 256 * 4);
  float* scaleB = (float*)take(1024 * 4);
  float* shiftB = (float*)take(1024 * 4);
  float* wpad   = (float*)take((size_t)256 * 260 * 4); // padded-weight scratch

  auto conv = [&](const float* Xin, float* Yout, const float* w, const float* g,
                  const float* b, int M, int Nc, int Kc) {
    const int KP = (Kc + 3) & ~3;
    const float* Wuse = w;
    if (KP != Kc) {
      const int tot = Nc * KP;
      pack_weights_kernel<<<(tot + 255) / 256, 256, 0, stream>>>(w, Kc, KP, wpad, tot);
      Wuse = wpad;
    }
    launch_gemm(Xin, Wuse, Yout, M, Nc, KP, stream);
    bn_stats_kernel<<<Nc, 256, 0, stream>>>(Yout, M, Nc, g, b, scaleB, shiftB);
    const int total = M * Nc;
    bn_relu_kernel<<<(total + 255) / 256, 256, 0, stream>>>(Yout, scaleB, shiftB, Nc, total);
  };

  // ---------------- Layer 1 ----------------
  {
    const size_t shm = (size_t)(4 * N1 + 256) * 4 + 256 * 4;
    fps_kernel<<<B, 256, shm, stream>>>(xyz_time, N1, 4, S1, fidx1);
  }
  {
    const int tot = B * S1 * 3;
    gather_xyz_kernel<<<(tot + 255) / 256, 256, 0, stream>>>(xyz_time, N1, 4, fidx1, S1, nxyz1, tot);
  }
  {
    const int tot = B * S1;
    query_ball_kernel<<<(tot + 255) / 256, 256, 0, stream>>>(xyz_time, N1, 4, nxyz1,
                                                             0.2f * 0.2f, NS1, gidx1, S1, tot);
  }
  build_points_kernel<<<(M1 + 255) / 256, 256, 0, stream>>>(
      xyz_time, N1, 4, xyz_time + 3, 4, 1, nxyz1, gidx1, S1, NS1, 4, X1, M1);

  conv(X1,   bufA, Wgt(1, 0), Gam(1, 0), Bet(1, 0), M1, 64, 4);
  conv(bufA, bufB, Wgt(1, 1), Gam(1, 1), Bet(1, 1), M1, 64, 64);
  conv(bufB, bufA, Wgt(1, 2), Gam(1, 2), Bet(1, 2), M1, 128, 64);
  {
    const int tot = B * S1 * 128;
    maxpool_kernel<<<(tot + 255) / 256, 256, 0, stream>>>(bufA, NS1, 128, feat1, tot);
  }

  // ---------------- Layer 2 ----------------
  {
    const size_t shm = (size_t)(4 * S1 + 256) * 4 + 256 * 4;
    fps_kernel<<<B, 256, shm, stream>>>(nxyz1, S1, 3, S2, fidx2);
  }
  {
    const int tot = B * S2 * 3;
    gather_xyz_kernel<<<(tot + 255) / 256, 256, 0, stream>>>(nxyz1, S1, 3, fidx2, S2, nxyz2, tot);
  }
  {
    const int tot = B * S2;
    query_ball_kernel<<<(tot + 255) / 256, 256, 0, stream>>>(nxyz1, S1, 3, nxyz2,
                                                             0.4f * 0.4f, NS2, gidx2, S2, tot);
  }
  // X2 = bufB, padded rows of 132 (131 real channels + 1 zero)
  build_points_kernel<<<(M2 + 255) / 256, 256, 0, stream>>>(
      nxyz1, S1, 3, feat1, 128, 128, nxyz2, gidx2, S2, NS2, 132, bufB, M2);

  conv(bufB, bufA, Wgt(2, 0), Gam(2, 0), Bet(2, 0), M2, 128, 131);
  conv(bufA, bufB, Wgt(2, 1), Gam(2, 1), Bet(2, 1), M2, 128, 128);
  conv(bufB, bufA, Wgt(2, 2), Gam(2, 2), Bet(2, 2), M2, 256, 128);
  {
    const int tot = M3 * 256;
    maxpool_kernel<<<(tot + 255) / 256, 256, 0, stream>>>(bufA, NS2, 256, feat2, tot);
  }

  // ---------------- Layer 3 (group_all) ----------------
  {
    // X3 = bufB, padded rows of 260 (259 real channels + 1 zero)
    const int tot = M3 * 260;
    concat_all_kernel<<<(tot + 255) / 256, 256, 0, stream>>>(nxyz2, feat2, 256, 260, bufB, tot);
  }
  conv(bufB, bufA, Wgt(3, 0), Gam(3, 0), Bet(3, 0), M3, 256, 259);
  conv(bufA, bufB, Wgt(3, 1), Gam(3, 1), Bet(3, 1), M3, 512, 256);
  conv(bufB, bufA, Wgt(3, 2), Gam(3, 2), Bet(3, 2), M3, 1024, 512);
  {
    const int tot = B * 1024;  // (8,1024) == (2,4,1024) flat
    maxpool_kernel<<<(tot + 255) / 256, 256, 0, stream>>>(bufA, S2, 1024, (float*)d_out, tot);
  }
}